// PointNet2FPModule_5007931867450
// MI455X (gfx1250) — compile-verified
//
#include <hip/hip_runtime.h>
#include <hip/hip_bf16.h>

typedef __attribute__((ext_vector_type(2))) float v2f;
typedef __attribute__((ext_vector_type(8))) float v8f;

namespace {
constexpr int B_   = 16;
constexpr int N_   = 4096;   // n (unknown points)
constexpr int M_   = 1024;   // m (known points)
constexpr int C1_  = 256;    // unknow_feats channels
constexpr int C2_  = 512;    // known_feats channels
constexpr int CH_  = 256;    // MLP hidden/out channels
constexpr float BN_EPS_ = 1e-5f;

constexpr int TM = 64;       // block tile M
constexpr int TN = 256;      // block tile N
constexpr int TK = 16;       // staged K per iteration
constexpr int LDA_PAD = 20;  // padded LDS stride for A tile (floats)
constexpr int LDB_PAD = 264; // padded LDS stride for B tile (floats)
}

// ---------------------------------------------------------------------------
// CDNA5 async global->LDS copy (ASYNCcnt-tracked, bypasses VGPRs).
// VDST operand = 32-bit LDS byte address; per the flat-aperture mapping
// (ISA 10.2) this equals the low 32 bits of the generic pointer to LDS.
// ---------------------------------------------------------------------------
__device__ __forceinline__ void async_copy_b128(const float* __restrict__ gsrc,
                                                float* ldst) {
  const unsigned           lds_addr = (unsigned)(size_t)ldst;
  const unsigned long long gaddr    = (unsigned long long)(size_t)gsrc;
  asm volatile("global_load_async_to_lds_b128 %0, %1, off"
               :: "v"(lds_addr), "v"(gaddr) : "memory");
}

__device__ __forceinline__ void wait_async0() {
#if __has_builtin(__builtin_amdgcn_s_wait_asynccnt)
  __builtin_amdgcn_s_wait_asynccnt(0);
#else
  asm volatile("s_wait_asynccnt 0x0" ::: "memory");
#endif
}

// ---------------------------------------------------------------------------
// Kernel 1: 3-NN search + inverse-distance weights
// grid (n/256, B), block 256 ; one thread per unknown point
// ---------------------------------------------------------------------------
__global__ __launch_bounds__(256)
void three_nn_weights(const float* __restrict__ unknown,   // [B, n, 3]
                      const float* __restrict__ known,     // [B, m, 3]
                      int*   __restrict__ idx_out,         // [B, n, 3]
                      float* __restrict__ w_out) {         // [B, n, 3]
  const int b = blockIdx.y;
  const int i = blockIdx.x * 256 + threadIdx.x;
  const float* u = unknown + ((size_t)b * N_ + i) * 3;
  const float ux = u[0], uy = u[1], uz = u[2];

  __shared__ float kx[256], ky[256], kz[256];

  float d0 = 1e30f, d1 = 1e30f, d2 = 1e30f;
  int   i0 = 0, i1 = 0, i2 = 0;

  for (int j0 = 0; j0 < M_; j0 += 256) {
    __syncthreads();
    const float* kp = known + ((size_t)b * M_ + j0 + threadIdx.x) * 3;
    kx[threadIdx.x] = kp[0];
    ky[threadIdx.x] = kp[1];
    kz[threadIdx.x] = kp[2];
    __syncthreads();
    #pragma unroll 4
    for (int jj = 0; jj < 256; ++jj) {
      const float dx = ux - kx[jj];
      const float dy = uy - ky[jj];
      const float dz = uz - kz[jj];
      const float d  = dx * dx + dy * dy + dz * dz;
      const int jg = j0 + jj;
      if (d < d2) {
        if (d < d1) {
          d2 = d1; i2 = i1;
          if (d < d0) { d1 = d0; i1 = i0; d0 = d; i0 = jg; }
          else        { d1 = d;  i1 = jg; }
        } else { d2 = d; i2 = jg; }
      }
    }
  }

  const float r0 = 1.0f / (d0 + 1e-8f);
  const float r1 = 1.0f / (d1 + 1e-8f);
  const float r2 = 1.0f / (d2 + 1e-8f);
  const float rs = 1.0f / (r0 + r1 + r2);

  const size_t base = ((size_t)b * N_ + i) * 3;
  idx_out[base + 0] = i0;
  idx_out[base + 1] = i1;
  idx_out[base + 2] = i2;
  w_out[base + 0] = r0 * rs;
  w_out[base + 1] = r1 * rs;
  w_out[base + 2] = r2 * rs;
}

// ---------------------------------------------------------------------------
// Kernel 2: weighted gather-interpolation -> interp [B, C2, n]
// known_feats (33.5 MB) is L2-resident (192 MB L2), so the gather is cheap.
// ---------------------------------------------------------------------------
__global__ __launch_bounds__(256)
void gather_interp(const float* __restrict__ kf,   // [B, C2, m]
                   const int*   __restrict__ idx,  // [B, n, 3]
                   const float* __restrict__ w,    // [B, n, 3]
                   float* __restrict__ out) {      // [B, C2, n]
  const int b = blockIdx.y;
  const int i = blockIdx.x * 256 + threadIdx.x;
  const size_t base = ((size_t)b * N_ + i) * 3;
  const int j0 = idx[base + 0], j1 = idx[base + 1], j2 = idx[base + 2];
  const float w0 = w[base + 0], w1 = w[base + 1], w2 = w[base + 2];

  const float* kfb = kf + (size_t)b * C2_ * M_;
  float* ob = out + (size_t)b * C2_ * N_ + i;
  #pragma unroll 4
  for (int c = 0; c < C2_; ++c) {
    const float* row = kfb + (size_t)c * M_;
    ob[(size_t)c * N_] = w0 * row[j0] + w1 * row[j1] + w2 * row[j2];
  }
}

// ---------------------------------------------------------------------------
// Kernel 3: fp32 WMMA GEMM   Y[b] = A(MxK) * X[b](KxN)   (pre-BN output)
// X split in K: rows [0,K0) from X0, rows [K0,K0+K1) from X1.
// Block: 256 threads = 8 waves (2 in M x 4 in N); block tile 64x256;
// wave tile 32x64 = 2x4 v_wmma_f32_16x16x4_f32 accumulators.
// Staging: double-buffered GLOBAL_LOAD_ASYNC_TO_LDS_B128 (ASYNCcnt).
// ---------------------------------------------------------------------------
__global__ __launch_bounds__(256)
void fp_wmma_gemm(const float* __restrict__ A,    // [M x K] row-major (shared over batch)
                  const float* __restrict__ X0,   // [B][K0 x N]
                  const float* __restrict__ X1,   // [B][K1 x N] (may be null iff K1==0)
                  float* __restrict__ Y,          // [B][M x N]
                  int M, int K0, int K1) {
  const int N = N_;
  const int K = K0 + K1;
  const int b  = blockIdx.z;
  const int m0 = blockIdx.y * TM;
  const int n0 = blockIdx.x * TN;
  const int t    = threadIdx.x;
  const int wave = t >> 5;
  const int lane = t & 31;
  const int my = wave >> 2;        // 0..1  (M position of wave)
  const int wx = wave & 3;         // 0..3  (N position of wave)
  const int l16 = lane & 15;
  const int hi  = lane >> 4;       // 0 or 1 (half-wave)

  __shared__ float sA[2][TM * LDA_PAD];  // 2 x  5.0 KB
  __shared__ float sB[2][TK * LDB_PAD];  // 2 x 16.5 KB

  v8f acc[2][4];
  #pragma unroll
  for (int mf = 0; mf < 2; ++mf)
    #pragma unroll
    for (int nf = 0; nf < 4; ++nf) {
      v8f z = {};
      acc[mf][nf] = z;
    }

  // A-tile loader mapping: one b128 per thread (64 rows x 16 k)
  const int aRow = t >> 2;
  const int aCol = (t & 3) << 2;
  const float* aSrc = A + (size_t)(m0 + aRow) * K + aCol;

  // stage K-tile starting at k0 into buffer buf (async, no VGPR round trip)
  auto stage = [&](int k0, int buf) {
    async_copy_b128(aSrc + k0, &sA[buf][aRow * LDA_PAD + aCol]);
    #pragma unroll
    for (int q = 0; q < 4; ++q) {
      const int f    = t + (q << 8);
      const int row  = f >> 6;         // 0..15
      const int c4   = (f & 63) << 2;  // 0..252
      const int krow = k0 + row;
      const float* src = (krow < K0)
          ? (X0 + ((size_t)b * K0 + krow) * N)
          : (X1 + ((size_t)b * K1 + (krow - K0)) * N);
      async_copy_b128(src + n0 + c4, &sB[buf][row * LDB_PAD + c4]);
    }
  };

  const int nT = K / TK;
  stage(0, 0);

  for (int kt = 0; kt < nT; ++kt) {
    const int cur = kt & 1;
    wait_async0();       // my async copies into sA/sB[cur] are complete
    __syncthreads();     // everyone's copies complete; buffer reuse safe

    if (kt + 1 < nT) stage((kt + 1) * TK, cur ^ 1);  // prefetch next tile

    const float* __restrict__ cA = sA[cur];
    const float* __restrict__ cB = sB[cur];

    #pragma unroll
    for (int kk = 0; kk < TK; kk += 4) {
      v2f afr[2];
      #pragma unroll
      for (int mf = 0; mf < 2; ++mf) {
        // A 16x4 layout: lanes 0-15 -> K = kk,kk+1 ; lanes 16-31 -> K = kk+2,kk+3
        const float* p = &cA[(my * 32 + mf * 16 + l16) * LDA_PAD + kk + 2 * hi];
        afr[mf].x = p[0];
        afr[mf].y = p[1];
      }
      #pragma unroll
      for (int nf = 0; nf < 4; ++nf) {
        const int off = wx * 64 + nf * 16 + l16;
        v2f bfr;
        bfr.x = cB[(kk + 2 * hi + 0) * LDB_PAD + off];
        bfr.y = cB[(kk + 2 * hi + 1) * LDB_PAD + off];
        #pragma unroll
        for (int mf = 0; mf < 2; ++mf) {
          acc[mf][nf] = __builtin_amdgcn_wmma_f32_16x16x4_f32(
              false, afr[mf], false, bfr, (short)0, acc[mf][nf], false, false);
        }
      }
    }
  }

  // ---- epilogue: C/D layout: VGPR r -> M=r (lanes 0-15) / M=r+8 (lanes 16-31)
  #pragma unroll
  for (int mf = 0; mf < 2; ++mf) {
    #pragma unroll
    for (int nf = 0; nf < 4; ++nf) {
      const int col  = n0 + wx * 64 + nf * 16 + l16;
      const int rowb = m0 + my * 32 + mf * 16 + hi * 8;
      float* dst = Y + ((size_t)b * M + rowb) * N + col;
      #pragma unroll
      for (int r = 0; r < 8; ++r)
        dst[(size_t)r * N] = acc[mf][nf][r];
    }
  }
}

// ---------------------------------------------------------------------------
// Kernel 4: per-channel training-mode BN statistics (mean, rsqrt(var+eps))
// grid = CH_, block 256; one block fully reduces one channel over B*n.
// ---------------------------------------------------------------------------
__global__ __launch_bounds__(256)
void bn_stats(const float* __restrict__ Y,      // [B, CH_, n]
              float* __restrict__ stats) {      // [2*CH_]: mean, rstd
  const int c = blockIdx.x;
  const int t = threadIdx.x;
  float s = 0.0f, s2 = 0.0f;
  for (int b = 0; b < B_; ++b) {
    const float* row = Y + ((size_t)b * CH_ + c) * N_;
    for (int i = t; i < N_; i += 256) {
      const float v = row[i];
      s += v;
      s2 += v * v;
    }
  }
  __shared__ float r1[256], r2[256];
  r1[t] = s;
  r2[t] = s2;
  __syncthreads();
  for (int off = 128; off > 0; off >>= 1) {
    if (t < off) {
      r1[t] += r1[t + off];
      r2[t] += r2[t + off];
    }
    __syncthreads();
  }
  if (t == 0) {
    const float inv = 1.0f / (float)(B_ * N_);
    const float mu  = r1[0] * inv;
    const float var = r2[0] * inv - mu * mu;
    stats[c]        = mu;
    stats[CH_ + c]  = rsqrtf(fmaxf(var, 0.0f) + BN_EPS_);
  }
}

// ---------------------------------------------------------------------------
// Kernel 5: BN apply + ReLU, in place on [B, CH_, n]
// ---------------------------------------------------------------------------
__global__ __launch_bounds__(256)
void bn_relu_inplace(float* __restrict__ Y,
                     const float* __restrict__ stats,
                     const float* __restrict__ gamma,
                     const float* __restrict__ beta) {
  const size_t idx = (size_t)blockIdx.x * 256 + threadIdx.x;
  const int c = (int)((idx >> 12) & (CH_ - 1));  // n = 4096 = 2^12, CH_ = 256
  const float mu   = stats[c];
  const float rstd = stats[CH_ + c];
  float v = Y[idx];
  v = gamma[c] * (v - mu) * rstd + beta[c];
  Y[idx] = fmaxf(v, 0.0f);
}

// ---------------------------------------------------------------------------
// Host-side launch
// ---------------------------------------------------------------------------
extern "C" void kernel_launch(void* const* d_in, const int* in_sizes, int n_in,
                              void* d_out, int out_size, void* d_ws, size_t ws_size,
                              hipStream_t stream) {
  (void)in_sizes; (void)n_in; (void)out_size; (void)ws_size;

  const float* unknown = (const float*)d_in[0];   // [B, n, 3]
  const float* known   = (const float*)d_in[1];   // [B, m, 3]
  const float* uf      = (const float*)d_in[2];   // [B, C1, n]
  const float* kf      = (const float*)d_in[3];   // [B, C2, m]
  const float* W1      = (const float*)d_in[4];   // [CH, CIN]
  const float* g1      = (const float*)d_in[5];
  const float* b1      = (const float*)d_in[6];
  const float* W2      = (const float*)d_in[7];   // [CH, CH]
  const float* g2      = (const float*)d_in[8];
  const float* b2      = (const float*)d_in[9];
  float* out = (float*)d_out;                     // [B, CH, n]

  // workspace carve-up
  char* ws = (char*)d_ws;
  int*   idx_buf = (int*)ws;      ws += (size_t)B_ * N_ * 3 * sizeof(int);
  float* w_buf   = (float*)ws;    ws += (size_t)B_ * N_ * 3 * sizeof(float);
  float* interp  = (float*)ws;    ws += (size_t)B_ * C2_ * N_ * sizeof(float); // 134 MB
  float* h_buf   = (float*)ws;    ws += (size_t)B_ * CH_ * N_ * sizeof(float); //  67 MB
  float* st1     = (float*)ws;    ws += 2 * CH_ * sizeof(float);
  float* st2     = (float*)ws;    ws += 2 * CH_ * sizeof(float);

  // 1) 3-NN + weights
  three_nn_weights<<<dim3(N_ / 256, B_), 256, 0, stream>>>(unknown, known, idx_buf, w_buf);

  // 2) gather interpolation -> interp [B, C2, n]
  gather_interp<<<dim3(N_ / 256, B_), 256, 0, stream>>>(kf, idx_buf, w_buf, interp);

  // 3) GEMM1: h = W1 * [interp ; unknow_feats]   (K = 512 + 256 = 768)
  fp_wmma_gemm<<<dim3(N_ / TN, CH_ / TM, B_), 256, 0, stream>>>(
      W1, interp, uf, h_buf, CH_, C2_, C1_);

  // 4) BN1 stats + apply + ReLU
  bn_stats<<<CH_, 256, 0, stream>>>(h_buf, st1);
  bn_relu_inplace<<<(B_ * CH_ * N_) / 256, 256, 0, stream>>>(h_buf, st1, g1, b1);

  // 5) GEMM2: out = W2 * h   (K = 256)
  fp_wmma_gemm<<<dim3(N_ / TN, CH_ / TM, B_), 256, 0, stream>>>(
      W2, h_buf, nullptr, out, CH_, CH_, 0);

  // 6) BN2 stats + apply + ReLU (in place on d_out)
  bn_stats<<<CH_, 256, 0, stream>>>(out, st2);
  bn_relu_inplace<<<(B_ * CH_ * N_) / 256, 256, 0, stream>>>(out, st2, g2, b2);
}